// PointNetSetAbstraction_55456617726251
// MI455X (gfx1250) — compile-verified
//
#include <hip/hip_runtime.h>
#include <math.h>

// ---------------- problem constants (fixed by reference) ----------------
#define PB   16          // batch
#define PN   4096        // points per batch
#define PD   64          // input feature channels
#define PS   1024        // npoint
#define PK   32          // nsample
#define PM   (PB * PS * PK)   // 524288 rows through the MLP
#define K0P  68          // 3 + 64 = 67 padded to multiple of 4
#define STATS_BLOCKS 1024

typedef float v2f __attribute__((ext_vector_type(2)));
typedef float v8f __attribute__((ext_vector_type(8)));

// ---------------- 1) gather new_xyz = xyz[fps_idx] ----------------
__global__ void gather_new_xyz_k(const float* __restrict__ xyz,
                                 const int* __restrict__ fps,
                                 float* __restrict__ out) {
    int i = blockIdx.x * blockDim.x + threadIdx.x;   // over B*S
    if (i >= PB * PS) return;
    int b = i / PS;
    int n = fps[i];
    const float* src = xyz + ((long)b * PN + n) * 3;
    float* dst = out + (long)i * 3;
    dst[0] = src[0]; dst[1] = src[1]; dst[2] = src[2];
}

// ---------------- 2) 32-NN: one thread per query, LDS point tiles ----------------
#define KNN_TPB 128
__global__ void knn_k(const float* __restrict__ xyz,
                      const int* __restrict__ fps,
                      int* __restrict__ idxout) {
    __shared__ float spx[KNN_TPB], spy[KNN_TPB], spz[KNN_TPB];
    const int b = blockIdx.y;
    const int s = blockIdx.x * KNN_TPB + threadIdx.x;   // S = 8*128 exactly
    const float* bxyz = xyz + (long)b * PN * 3;
    const int qn = fps[b * PS + s];
    const float qx = bxyz[qn * 3 + 0];
    const float qy = bxyz[qn * 3 + 1];
    const float qz = bxyz[qn * 3 + 2];

    float bd[PK];
    int   bi[PK];
#pragma unroll
    for (int k = 0; k < PK; ++k) { bd[k] = 3.4e38f; bi[k] = 0; }
    float worst = 3.4e38f;
    int   wpos = 0;

    for (int base = 0; base < PN; base += KNN_TPB) {
        __syncthreads();
        int p = base + threadIdx.x;
        spx[threadIdx.x] = bxyz[p * 3 + 0];
        spy[threadIdx.x] = bxyz[p * 3 + 1];
        spz[threadIdx.x] = bxyz[p * 3 + 2];
        __syncthreads();
        for (int j = 0; j < KNN_TPB; ++j) {
            float dx = spx[j] - qx, dy = spy[j] - qy, dz = spz[j] - qz;
            float d = dx * dx + dy * dy + dz * dz;
            if (d < worst) {
                bd[wpos] = d; bi[wpos] = base + j;
                worst = bd[0]; wpos = 0;
#pragma unroll
                for (int k = 1; k < PK; ++k)
                    if (bd[k] > worst) { worst = bd[k]; wpos = k; }
            }
        }
    }
    int* op = idxout + ((long)(b * PS + s)) * PK;
#pragma unroll
    for (int k = 0; k < PK; ++k) op[k] = bi[k];
}

// ---------------- 3) build F0 (M x 68): [xyz - center | points | 0 pad] ----------------
__global__ void build_features_k(const float* __restrict__ xyz,
                                 const float* __restrict__ points,
                                 const int* __restrict__ idx,
                                 const float* __restrict__ newxyz,
                                 float* __restrict__ F0) {
    long r = (long)blockIdx.x * blockDim.x + threadIdx.x;
    if (r >= (long)PM) return;
    long bs = r >> 5;                 // query index b*S+s
    int b = (int)(bs / PS);
    int n = idx[r];
    const float* src = xyz + ((long)b * PN + n) * 3;
    const float* ctr = newxyz + bs * 3;
    float* dst = F0 + r * K0P;
    dst[0] = src[0] - ctr[0];
    dst[1] = src[1] - ctr[1];
    dst[2] = src[2] - ctr[2];
    const float* prow = points + ((long)b * PN + n) * PD;
    for (int j = 0; j < PD; ++j) dst[3 + j] = prow[j];
    dst[67] = 0.0f;
}

// ---------------- 4) pad/transpose weights to K-major Bt[k][n] ----------------
__global__ void prep_weights_k(const float* __restrict__ w0,
                               const float* __restrict__ w1,
                               const float* __restrict__ w2,
                               float* __restrict__ Bt0,
                               float* __restrict__ Bt1,
                               float* __restrict__ Bt2) {
    int t = blockIdx.x * blockDim.x + threadIdx.x;
    int stride = gridDim.x * blockDim.x;
    for (int i = t; i < K0P * 64; i += stride) {       // 68 x 64, in_c = 67
        int k = i / 64, n = i % 64;
        Bt0[i] = (k < 67) ? w0[n * 67 + k] : 0.0f;
    }
    for (int i = t; i < 64 * 64; i += stride) {        // 64 x 64
        int k = i / 64, n = i % 64;
        Bt1[i] = w1[n * 64 + k];
    }
    for (int i = t; i < 64 * 128; i += stride) {       // 64 x 128
        int k = i / 128, n = i % 128;
        Bt2[i] = w2[n * 64 + k];
    }
}

// ---------------- 5) fp32 WMMA GEMM: C[M,N] = A[M,Kp] * Bt[Kp,N] + bias ----------------
// One wave computes a full 16xN strip (NT = N/16 accumulators) so the big A
// operand is read exactly once; B (tiny, L2-resident) is reused from cache.
template <int NT>
__global__ __launch_bounds__(256)
void gemm_wmma_f32_k(const float* __restrict__ A,
                     const float* __restrict__ Bt,
                     const float* __restrict__ bias,
                     float* __restrict__ C,
                     int M, int Kp) {
    constexpr int N = NT * 16;
    const int lane = threadIdx.x & 31;
    const int wave = threadIdx.x >> 5;
    int tm = blockIdx.x * (blockDim.x >> 5) + wave;     // wave-uniform
    if (tm >= (M >> 4)) return;                         // whole-wave exit: EXEC stays all-1

    // A 16x4 fragment layout: lane L (<16): K = k0,k0+1 ; lane L (>=16): K = k0+2,k0+3 ; M = L%16
    const int mrow = tm * 16 + (lane & 15);
    const int coll = lane & 15;                         // column within a 16-wide tile
    const int kb   = (lane >> 4) << 1;                  // 0 or 2

    v8f acc[NT];
#pragma unroll
    for (int t = 0; t < NT; ++t) acc[t] = (v8f){};

    const float* arow = A + (long)mrow * Kp;
    for (int k0 = 0; k0 < Kp; k0 += 4) {
        v2f a;
        a.x = arow[k0 + kb];
        a.y = arow[k0 + kb + 1];
        const float* b0p = Bt + (k0 + kb) * N + coll;
        const float* b1p = Bt + (k0 + kb + 1) * N + coll;
#pragma unroll
        for (int t = 0; t < NT; ++t) {
            v2f b;
            b.x = b0p[t * 16];
            b.y = b1p[t * 16];
            acc[t] = __builtin_amdgcn_wmma_f32_16x16x4_f32(
                         /*neg_a=*/false, a, /*neg_b=*/false, b,
                         /*c_mod=*/(short)0, acc[t], /*reuse_a=*/false, /*reuse_b=*/false);
        }
    }
    // D layout: VGPR v, lanes 0-15 -> M=v, N=lane ; lanes 16-31 -> M=v+8, N=lane-16
    const int rbase = tm * 16 + ((lane >> 4) << 3);
#pragma unroll
    for (int t = 0; t < NT; ++t) {
        float bv = bias[t * 16 + coll];
#pragma unroll
        for (int v = 0; v < 8; ++v) {
            C[(long)(rbase + v) * N + t * 16 + coll] = acc[t][v] + bv;
        }
    }
}

// ---------------- 6) BN stats: deterministic per-block partial sums ----------------
template <int N>
__global__ void bn_stats_k(const float* __restrict__ X,
                           float* __restrict__ partials, int M) {
    constexpr int TPB = 256;
    constexpr int RG = TPB / N;
    __shared__ float ls[2 * TPB];
    const int c = threadIdx.x & (N - 1);
    const int rg = threadIdx.x / N;
    float s = 0.0f, q = 0.0f;
    for (long row = (long)blockIdx.x * RG + rg; row < M; row += (long)gridDim.x * RG) {
        float v = X[row * N + c];
        s += v;
        q += v * v;
    }
    ls[threadIdx.x] = s;
    ls[TPB + threadIdx.x] = q;
    __syncthreads();
    if (threadIdx.x < N) {               // here c == threadIdx.x
        float ts = 0.0f, tq = 0.0f;
#pragma unroll
        for (int i = 0; i < RG; ++i) {
            ts += ls[i * N + c];
            tq += ls[TPB + i * N + c];
        }
        partials[(long)blockIdx.x * 2 * N + c]     = ts;
        partials[(long)blockIdx.x * 2 * N + N + c] = tq;
    }
}

template <int N>
__global__ void bn_reduce_k(const float* __restrict__ partials,
                            float* __restrict__ sums, int G) {
    int t = threadIdx.x;
    if (t < 2 * N) {
        float acc = 0.0f;
        for (int g = 0; g < G; ++g) acc += partials[(long)g * 2 * N + t];
        sums[t] = acc;                    // [0..N): sum, [N..2N): sumsq
    }
}

// ---------------- 7) BN(train) + ReLU, in place ----------------
template <int N>
__global__ void bn_apply_relu_k(float* __restrict__ X,
                                const float* __restrict__ sums,
                                const float* __restrict__ g,
                                const float* __restrict__ be, int M) {
    __shared__ float scale[N], shift[N];
    if (threadIdx.x < N) {
        int c = threadIdx.x;
        float inv = 1.0f / (float)M;
        float mean = sums[c] * inv;
        float var = sums[N + c] * inv - mean * mean;
        float sc = g[c] * rsqrtf(var + 1e-5f);
        scale[c] = sc;
        shift[c] = be[c] - mean * sc;
    }
    __syncthreads();
    long total = (long)M * N;
    for (long i = (long)blockIdx.x * blockDim.x + threadIdx.x; i < total;
         i += (long)gridDim.x * blockDim.x) {
        int c = (int)(i & (N - 1));
        float v = X[i] * scale[c] + shift[c];
        X[i] = v > 0.0f ? v : 0.0f;
    }
}

// ---------------- 8) max over nsample ----------------
__global__ void max_over_k_k(const float* __restrict__ A3, float* __restrict__ out) {
    int i = blockIdx.x * blockDim.x + threadIdx.x;    // over B*S*128
    if (i >= PB * PS * 128) return;
    int bs = i >> 7;
    int c = i & 127;
    const float* p = A3 + (long)bs * PK * 128 + c;
    float m = p[0];
#pragma unroll
    for (int k = 1; k < PK; ++k) m = fmaxf(m, p[k * 128]);
    out[i] = m;
}

// ---------------- host orchestration ----------------
extern "C" void kernel_launch(void* const* d_in, const int* in_sizes, int n_in,
                              void* d_out, int out_size, void* d_ws, size_t ws_size,
                              hipStream_t stream) {
    (void)in_sizes; (void)n_in; (void)out_size; (void)ws_size;

    const float* xyz    = (const float*)d_in[0];
    const float* points = (const float*)d_in[1];
    const int*   fps    = (const int*)d_in[2];
    const float* w0  = (const float*)d_in[3];
    const float* b0  = (const float*)d_in[4];
    const float* g0  = (const float*)d_in[5];
    const float* be0 = (const float*)d_in[6];
    const float* w1  = (const float*)d_in[7];
    const float* b1  = (const float*)d_in[8];
    const float* g1  = (const float*)d_in[9];
    const float* be1 = (const float*)d_in[10];
    const float* w2  = (const float*)d_in[11];
    const float* b2  = (const float*)d_in[12];
    const float* g2  = (const float*)d_in[13];
    const float* be2 = (const float*)d_in[14];

    float* out = (float*)d_out;
    float* new_xyz_out    = out;                       // B*S*3
    float* new_points_out = out + (long)PB * PS * 3;   // B*S*128

    // ---- workspace carve-up ----
    char* ws = (char*)d_ws;
    size_t off = 0;
    auto align256 = [](size_t x) { return (x + 255) & ~(size_t)255; };
    int* idxbuf = (int*)(ws + off);      off += (size_t)PB * PS * PK * 4;  off = align256(off);
    float* sums = (float*)(ws + off);    off += 512 * 4;                   off = align256(off);
    float* Bt0 = (float*)(ws + off);     off += (size_t)K0P * 64 * 4;      off = align256(off);
    float* Bt1 = (float*)(ws + off);     off += (size_t)64 * 64 * 4;       off = align256(off);
    float* Bt2 = (float*)(ws + off);     off += (size_t)64 * 128 * 4;      off = align256(off);
    float* partials = (float*)(ws + off);off += (size_t)STATS_BLOCKS * 256 * 4; off = align256(off);
    float* bufY = (float*)(ws + off);    off += (size_t)PM * K0P * 4;      off = align256(off); // F0 / A2
    float* bufX = (float*)(ws + off);    off += (size_t)PM * 128 * 4;                           // A1 / A3

    // 1) new_xyz
    gather_new_xyz_k<<<(PB * PS + 255) / 256, 256, 0, stream>>>(xyz, fps, new_xyz_out);
    // 2) knn
    knn_k<<<dim3(PS / KNN_TPB, PB), KNN_TPB, 0, stream>>>(xyz, fps, idxbuf);
    // 3) F0 (reads new_xyz from d_out; same stream => ordered)
    build_features_k<<<(PM + 255) / 256, 256, 0, stream>>>(xyz, points, idxbuf, new_xyz_out, bufY);
    // 4) weights
    prep_weights_k<<<32, 256, 0, stream>>>(w0, w1, w2, Bt0, Bt1, Bt2);

    const int wavesPerBlk = 256 / 32;
    const int rowTiles = PM / 16;                       // 32768 waves per GEMM
    const int gemmBlocks = (rowTiles + wavesPerBlk - 1) / wavesPerBlk;

    // ---- layer 1: (M x 68) * (68 x 64) -> bufX ----
    gemm_wmma_f32_k<4><<<gemmBlocks, 256, 0, stream>>>(bufY, Bt0, b0, bufX, PM, K0P);
    bn_stats_k<64><<<STATS_BLOCKS, 256, 0, stream>>>(bufX, partials, PM);
    bn_reduce_k<64><<<1, 128, 0, stream>>>(partials, sums, STATS_BLOCKS);
    bn_apply_relu_k<64><<<4096, 256, 0, stream>>>(bufX, sums, g0, be0, PM);

    // ---- layer 2: (M x 64) * (64 x 64) -> bufY ----
    gemm_wmma_f32_k<4><<<gemmBlocks, 256, 0, stream>>>(bufX, Bt1, b1, bufY, PM, 64);
    bn_stats_k<64><<<STATS_BLOCKS, 256, 0, stream>>>(bufY, partials, PM);
    bn_reduce_k<64><<<1, 128, 0, stream>>>(partials, sums, STATS_BLOCKS);
    bn_apply_relu_k<64><<<4096, 256, 0, stream>>>(bufY, sums, g1, be1, PM);

    // ---- layer 3: (M x 64) * (64 x 128) -> bufX ----
    gemm_wmma_f32_k<8><<<gemmBlocks, 256, 0, stream>>>(bufY, Bt2, b2, bufX, PM, 64);
    bn_stats_k<128><<<STATS_BLOCKS, 256, 0, stream>>>(bufX, partials, PM);
    bn_reduce_k<128><<<1, 256, 0, stream>>>(partials, sums, STATS_BLOCKS);
    bn_apply_relu_k<128><<<4096, 256, 0, stream>>>(bufX, sums, g2, be2, PM);

    // ---- max over K ----
    max_over_k_k<<<(PB * PS * 128 + 255) / 256, 256, 0, stream>>>(bufX, new_points_out);
}